// CrossEntropy_29222957482462
// MI455X (gfx1250) — compile-verified
//
#include <hip/hip_runtime.h>
#include <stdint.h>

// CrossEntropy-like row reduction on MI455X (gfx1250, wave32).
// B=16384 rows, C=4096 cols. Bandwidth-bound: ~512 MiB traffic => ~23us floor
// at 23.3 TB/s. Targets are streamed via CDNA5 async global->LDS DMA
// (GLOBAL_LOAD_ASYNC_TO_LDS_B128 / ASYNCcnt) overlapped with softmax pass 1.

#define ROWS   16384
#define COLS   4096
#define TPB    256
#define NWAVES (TPB / 32)
#define CHUNKS (COLS / (TPB * 4))   // 4 float4/int4 chunks per thread

#define AS1 __attribute__((address_space(1)))
#define AS3 __attribute__((address_space(3)))

typedef __attribute__((ext_vector_type(4))) int v4i_t;

#if defined(__HIP_DEVICE_COMPILE__)
#  if __has_builtin(__builtin_amdgcn_global_load_async_to_lds_b128)
#    define USE_ASYNC_BUILTIN 1
#  endif
#  if __has_builtin(__builtin_amdgcn_s_wait_asynccnt)
#    define USE_WAIT_BUILTIN 1
#  endif
#endif

__device__ __forceinline__ void async_copy_b128_to_lds(const void* gsrc, void* ldst) {
#if defined(USE_ASYNC_BUILTIN)
  // Prototype (from clang diagnostic): (v4i AS1* gsrc, v4i AS3* lds_dst, Ii offset, Ii cpol)
  __builtin_amdgcn_global_load_async_to_lds_b128((AS1 v4i_t*)gsrc, (AS3 v4i_t*)ldst, 0, 0);
#else
  // Fallback: raw CDNA5 mnemonic. vdst = 32-bit LDS byte offset, vaddr = 64-bit
  // global address (GV mode).
  unsigned lds_off = (unsigned)(uintptr_t)(AS3 void*)ldst;
  unsigned long long ga = (unsigned long long)(uintptr_t)gsrc;
  asm volatile("global_load_async_to_lds_b128 %0, %1, off"
               :: "v"(lds_off), "v"(ga) : "memory");
#endif
}

__device__ __forceinline__ void wait_async_zero() {
#if defined(USE_WAIT_BUILTIN)
  __builtin_amdgcn_s_wait_asynccnt(0);
#else
  asm volatile("s_wait_asynccnt 0" ::: "memory");
#endif
}

__global__ __launch_bounds__(TPB) void ce_rows_kernel(const float* __restrict__ x,
                                                      const int* __restrict__ tgt,
                                                      float* __restrict__ rowloss) {
  __shared__ int   s_tgt[COLS];        // 16 KB staged target row
  __shared__ float s_m[NWAVES];
  __shared__ float s_s[NWAVES];
  __shared__ float s_a[NWAVES];
  __shared__ float s_b[NWAVES];

  const int    row  = blockIdx.x;
  const int    t    = threadIdx.x;
  const size_t base = (size_t)row * COLS;
  const float4* xr  = (const float4*)(x + base);
  const int4*   tr  = (const int4*)(tgt + base);
  int4* s_tgt4      = (int4*)s_tgt;

  // Kick off async DMA of the target row into LDS; overlaps with pass 1.
#pragma unroll
  for (int k = 0; k < CHUNKS; ++k) {
    const int c = k * TPB + t;
    async_copy_b128_to_lds(tr + c, s_tgt4 + c);
  }

  // Pass 1a: load x row into registers (coalesced float4), local max.
  float4 v[CHUNKS];
#pragma unroll
  for (int k = 0; k < CHUNKS; ++k) v[k] = xr[k * TPB + t];

  float m = -3.402823466e38f;
#pragma unroll
  for (int k = 0; k < CHUNKS; ++k)
    m = fmaxf(m, fmaxf(fmaxf(v[k].x, v[k].y), fmaxf(v[k].z, v[k].w)));
#pragma unroll
  for (int o = 16; o > 0; o >>= 1) m = fmaxf(m, __shfl_xor(m, o, 32));
  if ((t & 31) == 0) s_m[t >> 5] = m;
  __syncthreads();
  m = s_m[0];
#pragma unroll
  for (int w = 1; w < NWAVES; ++w) m = fmaxf(m, s_m[w]);

  // Pass 1b: e = exp(x - max) kept in registers; block-reduce denominator.
  float4 e[CHUNKS];
  float sum = 0.0f;
#pragma unroll
  for (int k = 0; k < CHUNKS; ++k) {
    e[k].x = __expf(v[k].x - m);
    e[k].y = __expf(v[k].y - m);
    e[k].z = __expf(v[k].z - m);
    e[k].w = __expf(v[k].w - m);
    sum += (e[k].x + e[k].y) + (e[k].z + e[k].w);
  }
#pragma unroll
  for (int o = 16; o > 0; o >>= 1) sum += __shfl_xor(sum, o, 32);
  if ((t & 31) == 0) s_s[t >> 5] = sum;
  __syncthreads();
  sum = 0.0f;
#pragma unroll
  for (int w = 0; w < NWAVES; ++w) sum += s_s[w];
  const float inv = 1.0f / sum;

  // Pass 2: p = e*inv; accumulate sum(exp(p)) and target-masked sum(p).
  // Each thread reads exactly the LDS int4s its own wave DMA'd -> a per-wave
  // ASYNCcnt wait is sufficient.
  wait_async_zero();

  float a = 0.0f, bm = 0.0f;
#pragma unroll
  for (int k = 0; k < CHUNKS; ++k) {
    const int  c  = k * TPB + t;
    const int4 tg = s_tgt4[c];
    const float px = e[k].x * inv, py = e[k].y * inv;
    const float pz = e[k].z * inv, pw = e[k].w * inv;
    a  += (__expf(px) + __expf(py)) + (__expf(pz) + __expf(pw));
    bm += (tg.x == 1 ? px : 0.0f) + (tg.y == 1 ? py : 0.0f) +
          (tg.z == 1 ? pz : 0.0f) + (tg.w == 1 ? pw : 0.0f);
  }
#pragma unroll
  for (int o = 16; o > 0; o >>= 1) {
    a  += __shfl_xor(a,  o, 32);
    bm += __shfl_xor(bm, o, 32);
  }
  if ((t & 31) == 0) { s_a[t >> 5] = a; s_b[t >> 5] = bm; }
  __syncthreads();
  if (t == 0) {
    float A = 0.0f, Bm = 0.0f;
#pragma unroll
    for (int w = 0; w < NWAVES; ++w) { A += s_a[w]; Bm += s_b[w]; }
    rowloss[row] = __logf(A) - Bm;
  }
}

// Deterministic final mean over 16384 row losses (f64 accumulate, trivial cost).
__global__ __launch_bounds__(256) void ce_final_kernel(const float* __restrict__ rowloss,
                                                       float* __restrict__ out) {
  __shared__ double sd[256];
  double acc = 0.0;
  for (int i = threadIdx.x; i < ROWS; i += 256) acc += (double)rowloss[i];
  sd[threadIdx.x] = acc;
  __syncthreads();
  for (int s = 128; s > 0; s >>= 1) {
    if (threadIdx.x < s) sd[threadIdx.x] += sd[threadIdx.x + s];
    __syncthreads();
  }
  if (threadIdx.x == 0) out[0] = (float)(sd[0] * (1.0 / (double)ROWS));
}

extern "C" void kernel_launch(void* const* d_in, const int* in_sizes, int n_in,
                              void* d_out, int out_size, void* d_ws, size_t ws_size,
                              hipStream_t stream) {
  (void)in_sizes; (void)n_in; (void)out_size; (void)ws_size;
  const float* x   = (const float*)d_in[0];  // output: [B,C] f32
  const int*   tgt = (const int*)d_in[1];    // target: [B,C] i32 (0/1)
  float* rowloss   = (float*)d_ws;           // 16384 floats scratch (64 KB)
  float* out       = (float*)d_out;          // scalar f32

  ce_rows_kernel<<<ROWS, TPB, 0, stream>>>(x, tgt, rowloss);
  ce_final_kernel<<<1, 256, 0, stream>>>(rowloss, out);
}